// GCNEmbedderMulti_16896401343161
// MI455X (gfx1250) — compile-verified
//
#include <hip/hip_runtime.h>

// ---------------------------------------------------------------------------
// GCN (3-layer, 128ch) for MI455X / gfx1250.
// GEMM uses V_WMMA_F32_16X16X4_F32 (fp32 A/B/C -> matches fp32 reference
// exactly; GEMM is ~5 GFLOP total, negligible vs the L2-resident edge
// gather/scatter which dominates at ~800MB of L2 traffic per layer).
// ---------------------------------------------------------------------------

typedef float v2f __attribute__((ext_vector_type(2)));
typedef float v8f __attribute__((ext_vector_type(8)));

#define CH 128
#define ASTRIDE 132   // padded LDS row stride: 132 mod 64 = 4 -> conflict-free

// ---------------- degree / normalization -----------------------------------

__global__ __launch_bounds__(256) void k_init_deg(float* deg, int n) {
  int i = blockIdx.x * 256 + threadIdx.x;
  if (i < n) deg[i] = 1.0f;                 // self-loop contributes 1
}

__global__ __launch_bounds__(256) void k_count_deg(const int* __restrict__ dst,
                                                   float* deg, int e) {
  int i = blockIdx.x * 256 + threadIdx.x;
  if (i < e) atomicAdd(&deg[dst[i]], 1.0f);
}

__global__ __launch_bounds__(256) void k_dinv(float* deg, int n) {
  int i = blockIdx.x * 256 + threadIdx.x;
  if (i < n) deg[i] = rsqrtf(deg[i]);       // deg >= 1 always (self-loop)
}

// ---------------- GEMM: T = A @ W via v_wmma_f32_16x16x4_f32 ----------------
// Block = 256 threads = 8 waves. Block owns one 16-row tile; wave w owns
// column tile w (cols w*16..w*16+15). K = 128 -> 32 wmma steps of K=4,
// fully unrolled so the scheduler can pipeline ds/global loads vs wmma.

__global__ __launch_bounds__(256) void k_gemm_wmma(const float* __restrict__ A,
                                                   const float* __restrict__ W,
                                                   float* __restrict__ T,
                                                   int nRows) {
  __shared__ float Alds[16 * ASTRIDE];      // 16 x 128 tile, padded rows
  const int tid  = threadIdx.x;
  const int row0 = blockIdx.x * 16;

  // Coalesced stage of the A tile (2048 floats / 256 threads = 8 each).
  for (int idx = tid; idx < 16 * CH; idx += 256) {
    int r = idx >> 7;                       // / 128
    int c = idx & (CH - 1);
    int gr = row0 + r;
    Alds[r * ASTRIDE + c] = (gr < nRows) ? A[(size_t)gr * CH + c] : 0.0f;
  }
  __syncthreads();

  const int lane  = tid & 31;
  const int wave  = tid >> 5;               // 0..7 = column tile
  const int col   = (wave << 4) + (lane & 15);
  const int khalf = (lane >> 4) << 1;       // lanes 0-15 -> K+0/1, 16-31 -> K+2/3
  const float* Arow = &Alds[(lane & 15) * ASTRIDE];   // M = lane % 16

  v8f acc = {0.f, 0.f, 0.f, 0.f, 0.f, 0.f, 0.f, 0.f};

#pragma unroll
  for (int kk = 0; kk < CH; kk += 4) {
    const int k0 = kk + khalf;
    v2f a; a.x = Arow[k0];           a.y = Arow[k0 + 1];            // A 16x4 frag
    v2f b; b.x = W[k0 * CH + col];   b.y = W[(k0 + 1) * CH + col];  // B 4x16 frag
    acc = __builtin_amdgcn_wmma_f32_16x16x4_f32(
        /*neg_a=*/false, a, /*neg_b=*/false, b,
        /*c_mod=*/(short)0, acc, /*reuse_a=*/false, /*reuse_b=*/false);
  }

  // C/D layout: VGPR v -> row v (lanes 0-15) or v+8 (lanes 16-31), N = lane%16.
  // Fast path: whole tile in range (always true when nRows % 16 == 0) ->
  // 8 unconditional stores, no per-element exec masking.
  const int rbase = row0 + ((lane >> 4) << 3);
  float* __restrict__ Tp = T + (size_t)rbase * CH + col;
  if (rbase + 7 < nRows) {
#pragma unroll
    for (int v = 0; v < 8; ++v) Tp[(size_t)v * CH] = acc[v];
  } else {
#pragma unroll
    for (int v = 0; v < 8; ++v) {
      if (rbase + v < nRows) Tp[(size_t)v * CH] = acc[v];
    }
  }
}

// ---------------- self-loop init: agg = t * dinv^2 --------------------------

__global__ __launch_bounds__(256) void k_self_init(const float* __restrict__ t,
                                                   const float* __restrict__ dinv,
                                                   float* __restrict__ agg,
                                                   int total) {
  int idx = blockIdx.x * 256 + threadIdx.x;
  if (idx < total) {
    float di = dinv[idx >> 7];
    agg[idx] = t[idx] * di * di;
  }
}

// ---------------- edge scatter: one warp per edge ---------------------------
// Lane owns 4 channels -> coalesced float4 gather + 4 float atomics.

__global__ __launch_bounds__(256) void k_scatter(const float* __restrict__ t,
                                                 const int* __restrict__ src,
                                                 const int* __restrict__ dst,
                                                 const float* __restrict__ dinv,
                                                 float* __restrict__ agg,
                                                 int e) {
  int g    = blockIdx.x * 256 + threadIdx.x;
  int edge = g >> 5;
  int lane = g & 31;
  if (edge >= e) return;
  int s = src[edge];
  int d = dst[edge];
  float w = dinv[s] * dinv[d];
  const float4 v = *(const float4*)(t + (size_t)s * CH + lane * 4);
  float* ap = agg + (size_t)d * CH + lane * 4;
  atomicAdd(ap + 0, v.x * w);
  atomicAdd(ap + 1, v.y * w);
  atomicAdd(ap + 2, v.z * w);
  atomicAdd(ap + 3, v.w * w);
}

// ---------------- bias + optional ReLU --------------------------------------

__global__ __launch_bounds__(256) void k_bias_relu(const float* __restrict__ agg,
                                                   const float* __restrict__ b,
                                                   float* __restrict__ out,
                                                   int total, int doRelu) {
  int idx = blockIdx.x * 256 + threadIdx.x;
  if (idx < total) {
    float v = agg[idx] + b[idx & (CH - 1)];
    out[idx] = doRelu ? fmaxf(v, 0.0f) : v;
  }
}

// ---------------- column mean (+ bias fused): out[c] = mean(agg[:,c]) + b[c]

__global__ __launch_bounds__(256) void k_mean(const float* __restrict__ agg,
                                              const float* __restrict__ b,
                                              float* __restrict__ out, int n) {
  __shared__ float red[256];
  int c = blockIdx.x;                        // 0..127
  float s = 0.0f;
  for (int i = threadIdx.x; i < n; i += 256) s += agg[(size_t)i * CH + c];
  red[threadIdx.x] = s;
  __syncthreads();
  for (int off = 128; off > 0; off >>= 1) {
    if (threadIdx.x < off) red[threadIdx.x] += red[threadIdx.x + off];
    __syncthreads();
  }
  if (threadIdx.x == 0) out[c] = red[0] / (float)n + b[c];
}

// ---------------------------------------------------------------------------

extern "C" void kernel_launch(void* const* d_in, const int* in_sizes, int n_in,
                              void* d_out, int out_size, void* d_ws, size_t ws_size,
                              hipStream_t stream) {
  const float* x  = (const float*)d_in[0];
  const int*   ei = (const int*)d_in[1];
  const float* W1 = (const float*)d_in[2];
  const float* b1 = (const float*)d_in[3];
  const float* W2 = (const float*)d_in[4];
  const float* b2 = (const float*)d_in[5];
  const float* W3 = (const float*)d_in[6];
  const float* b3 = (const float*)d_in[7];
  float* outp = (float*)d_out;

  const int N = in_sizes[0] / CH;            // 50000
  const int E = in_sizes[1] / 2;             // 800000
  const int* src = ei;
  const int* dst = ei + E;

  const size_t NC = (size_t)N * CH;
  float* ws   = (float*)d_ws;
  size_t dOff = ((size_t)N + 64) & ~(size_t)63;  // aligned past dinv
  float* dinv = ws;
  float* t    = ws + dOff;
  float* agg  = t + NC;
  float* h    = agg + NC;

  const int gN   = (N + 255) / 256;
  const int gE   = (E + 255) / 256;
  const int gNC  = (int)((NC + 255) / 256);
  const int gEw  = (int)(((size_t)E * 32 + 255) / 256);  // warp per edge
  const int gTil = (N + 15) / 16;

  // --- normalization coefficients ---
  k_init_deg <<<gN, 256, 0, stream>>>(dinv, N);
  k_count_deg<<<gE, 256, 0, stream>>>(dst, dinv, E);
  k_dinv     <<<gN, 256, 0, stream>>>(dinv, N);

  // --- layer 1 ---
  k_gemm_wmma<<<gTil, 256, 0, stream>>>(x, W1, t, N);
  k_self_init<<<gNC, 256, 0, stream>>>(t, dinv, agg, (int)NC);
  k_scatter  <<<gEw, 256, 0, stream>>>(t, src, dst, dinv, agg, E);
  k_bias_relu<<<gNC, 256, 0, stream>>>(agg, b1, h, (int)NC, 1);

  // --- layer 2 ---
  k_gemm_wmma<<<gTil, 256, 0, stream>>>(h, W2, t, N);
  k_self_init<<<gNC, 256, 0, stream>>>(t, dinv, agg, (int)NC);
  k_scatter  <<<gEw, 256, 0, stream>>>(t, src, dst, dinv, agg, E);
  k_bias_relu<<<gNC, 256, 0, stream>>>(agg, b2, h, (int)NC, 1);

  // --- layer 3 (bias fused into mean, no ReLU) ---
  k_gemm_wmma<<<gTil, 256, 0, stream>>>(h, W3, t, N);
  k_self_init<<<gNC, 256, 0, stream>>>(t, dinv, agg, (int)NC);
  k_scatter  <<<gEw, 256, 0, stream>>>(t, src, dst, dinv, agg, E);
  k_mean     <<<CH, 256, 0, stream>>>(agg, b3, outp, N);
}